// MaskedLoss_68152541053394
// MI455X (gfx1250) — compile-verified
//
#include <hip/hip_runtime.h>

typedef __attribute__((ext_vector_type(2))) float v2f;
typedef __attribute__((ext_vector_type(8))) float v8f;

#define BLK   256
#define GRIDX 432   // 432*256*16 == 192^3/4 exactly (float4 elems per batch)

// ---------------------------------------------------------------------------
// Kernel A: zero the 64 global bins (sums[4][8] then counts[4][8]) in d_ws.
// ---------------------------------------------------------------------------
__global__ void zero_bins(float* __restrict__ ws) {
    int t = threadIdx.x;
    if (t < 64) ws[t] = 0.0f;
}

// ---------------------------------------------------------------------------
// Kernel B: streaming segmented reduction.
//   grid = (GRIDX, 4 batches), block = 256 (8 waves, wave32).
//   Each lane owns 8 private LDS bins; ds_add_f32 (atomicAdd on shared,
//   result unused -> no-return LDS atomic) does the binning in 1 DS op.
//   ~6 VALU + 2 DS per element << 14.6us HBM floor -> stays memory-bound.
// ---------------------------------------------------------------------------
__global__ __launch_bounds__(BLK) void masked_mse_partial(
    const float* __restrict__ in1, const float* __restrict__ tgt,
    const int*  __restrict__ mask, float* __restrict__ ws,
    int nv4_per_batch, long long n_per_batch)
{
    __shared__ float lds_s[BLK * 8];   // per-thread sum bins
    __shared__ float lds_c[BLK * 8];   // per-thread count bins
    __shared__ float ps[64], pc[64];   // partial fold (8 chunks x 8 ids)

    const int tid = threadIdx.x;
    const int b  = blockIdx.y;

    #pragma unroll
    for (int i = 0; i < 8; ++i) {
        lds_s[tid * 8 + i] = 0.0f;
        lds_c[tid * 8 + i] = 0.0f;
    }
    __syncthreads();

    const float4* __restrict__ a4 = (const float4*)(in1  + (size_t)b * n_per_batch);
    const float4* __restrict__ t4 = (const float4*)(tgt  + (size_t)b * n_per_batch);
    const int4*   __restrict__ m4 = (const int4*)  (mask + (size_t)b * n_per_batch);

    const int stride = GRIDX * BLK;
    float* const my_s = &lds_s[tid * 8];
    float* const my_c = &lds_c[tid * 8];

    for (int i = blockIdx.x * BLK + tid; i < nv4_per_batch; i += stride) {
        // speculative prefetch of next tile (gfx1250 global_prefetch_b8;
        // OOB translation failures are silently dropped)
        __builtin_prefetch(&a4[i + stride], 0, 0);
        __builtin_prefetch(&t4[i + stride], 0, 0);
        __builtin_prefetch(&m4[i + stride], 0, 0);

        float4 x = a4[i];
        float4 y = t4[i];
        int4   m = m4[i];

        float d;
        d = x.x - y.x; atomicAdd(&my_s[m.x & 7], d * d); atomicAdd(&my_c[m.x & 7], 1.0f);
        d = x.y - y.y; atomicAdd(&my_s[m.y & 7], d * d); atomicAdd(&my_c[m.y & 7], 1.0f);
        d = x.z - y.z; atomicAdd(&my_s[m.z & 7], d * d); atomicAdd(&my_c[m.z & 7], 1.0f);
        d = x.w - y.w; atomicAdd(&my_s[m.w & 7], d * d); atomicAdd(&my_c[m.w & 7], 1.0f);
    }
    __syncthreads();

    // fold 256 rows -> 8 chunk-partials per id -> 1 per id
    if (tid < 64) {
        const int id = tid & 7, chunk = tid >> 3;
        float s = 0.0f, c = 0.0f;
        const int base = chunk * 32;
        #pragma unroll 4
        for (int t = base; t < base + 32; ++t) {
            s += lds_s[t * 8 + id];
            c += lds_c[t * 8 + id];
        }
        ps[tid] = s;
        pc[tid] = c;
    }
    __syncthreads();

    if (tid < 8) {
        float s = 0.0f, c = 0.0f;
        #pragma unroll
        for (int ch = 0; ch < 8; ++ch) {
            s += ps[ch * 8 + tid];
            c += pc[ch * 8 + tid];
        }
        atomicAdd(&ws[b * 8 + tid],      s);  // sums[b][id]
        atomicAdd(&ws[32 + b * 8 + tid], c);  // counts[b][id]
    }
}

// ---------------------------------------------------------------------------
// Kernel C: finalize with an exact f32 WMMA reduction.
//   28 per-(batch,id!=0) means are placed into A of V_WMMA_F32_16X16X4_F32
//   (16x4 f32 slots, zero padded); B = all-ones 4x16. D row-sums are exact
//   f32 FMAs, so this is a numerically exact use of the matrix pipe.
//   A layout (32-bit 16x4): lanes 0-15 hold {K=0,K=1}, lanes 16-31 {K=2,K=3}.
//   slot s = k*16 + m  ->  lane L supplies s_x = (L<16 ? L : L+16), s_y = s_x+16.
// ---------------------------------------------------------------------------
__global__ void masked_mse_finalize(const float* __restrict__ ws,
                                    float* __restrict__ out)
{
    const int lane = threadIdx.x;            // 32 threads = 1 wave, EXEC all 1s
    const float* sums = ws;
    const float* cnts = ws + 32;

    float ax = 0.0f, ay = 0.0f;
    const int sx = (lane < 16) ? lane : lane + 16;
    if (sx < 28) {
        const int bb = sx / 7, id = 1 + sx % 7;
        const float c = cnts[bb * 8 + id];
        ax = (c > 0.0f) ? sums[bb * 8 + id] / c : 0.0f;
    }
    const int sy = sx + 16;
    if (sy < 28) {
        const int bb = sy / 7, id = 1 + sy % 7;
        const float c = cnts[bb * 8 + id];
        ay = (c > 0.0f) ? sums[bb * 8 + id] / c : 0.0f;
    }

    v2f a  = {ax, ay};
    v2f bv = {1.0f, 1.0f};   // every B element = 1 -> D[m][n] = sum_k A[m][k]
    v8f c  = {};
    c = __builtin_amdgcn_wmma_f32_16x16x4_f32(
            /*neg_a=*/false, a, /*neg_b=*/false, bv,
            /*c_mod=*/(short)0, c, /*reuse_a=*/false, /*reuse_b=*/false);

    // lane 0 holds row-sums m=0..7, lane 16 holds m=8..15 (n=0 column)
    float t = c[0] + c[1] + c[2] + c[3] + c[4] + c[5] + c[6] + c[7];
    float total = __shfl(t, 0, 32) + __shfl(t, 16, 32);
    if (lane == 0) out[0] = total * 0.25f;   // divide by B=4
}

// ---------------------------------------------------------------------------
extern "C" void kernel_launch(void* const* d_in, const int* in_sizes, int n_in,
                              void* d_out, int out_size, void* d_ws, size_t ws_size,
                              hipStream_t stream)
{
    const float* in1  = (const float*)d_in[0];
    const float* tgt  = (const float*)d_in[1];
    const int*   mask = (const int*)d_in[2];
    float* ws  = (float*)d_ws;
    float* out = (float*)d_out;

    const long long total       = (long long)in_sizes[0];   // 4*192^3
    const long long n_per_batch = total / 4;                 // 192^3
    const int       nv4         = (int)(n_per_batch / 4);    // float4 count

    zero_bins<<<1, 64, 0, stream>>>(ws);

    dim3 grid(GRIDX, 4);
    masked_mse_partial<<<grid, BLK, 0, stream>>>(in1, tgt, mask, ws,
                                                 nv4, n_per_batch);

    masked_mse_finalize<<<1, 32, 0, stream>>>(ws, out);
}